// ClusterAttention_68384469287496
// MI455X (gfx1250) — compile-verified
//
#include <hip/hip_runtime.h>
#include <stdint.h>

typedef float v2f __attribute__((ext_vector_type(2)));
typedef float v4f __attribute__((ext_vector_type(4)));
typedef float v8f __attribute__((ext_vector_type(8)));

#define C128   128
#define NBUCK  8192
#define GSCALE 0.17677669529663687f   // (128/4)^-0.5

// ---------------- WMMA helper: D = A(16x4) x B(4x16) + C, full fp32 ----------
static __device__ __forceinline__ v8f wmma4(v2f a, v2f b, v8f c) {
  return __builtin_amdgcn_wmma_f32_16x16x4_f32(false, a, false, b, (short)0, c,
                                               false, false);
}

// order-preserving float<->uint mapping for atomic max
static __device__ __forceinline__ unsigned fenc(float f) {
  unsigned b = __float_as_uint(f);
  return (b & 0x80000000u) ? ~b : (b | 0x80000000u);
}
static __device__ __forceinline__ float fdec(unsigned e) {
  unsigned b = (e & 0x80000000u) ? (e ^ 0x80000000u) : ~e;
  return __uint_as_float(b);
}

// ---------------- kernel 0: zero the stats region ----------------------------
__global__ void zero_kernel(unsigned* __restrict__ p, size_t n) {
  size_t i = (size_t)blockIdx.x * 256 + threadIdx.x;
  if (i < n) p[i] = 0u;
}

// ---------------- kernel 1: bucket counts ------------------------------------
__global__ void counts_kernel(const int* __restrict__ cids,
                              unsigned* __restrict__ counts, int n) {
  int i = blockIdx.x * 256 + threadIdx.x;
  if (i < n) atomicAdd(&counts[cids[i]], 1u);
}

// ---------------- kernel 2: repack weight (K x 128) into B-fragment order ----
// GEMM reads float2 at index (nt*(K/4)+ks)*32 + lane; element j of the pair is
// W[(4*ks + 2*h + j)*128 + nt*16 + r]  with h=lane>>4, r=lane&15.
__global__ void repack_kernel(const float* __restrict__ src,
                              float* __restrict__ dst, int K) {
  int idx = blockIdx.x * 256 + threadIdx.x;
  if (idx >= K * C128) return;
  int j    = idx & 1;
  int pair = idx >> 1;
  int lane = pair & 31;
  int t    = pair >> 5;
  int kq   = K >> 2;
  int ks   = t % kq;
  int nt   = t / kq;
  int h = lane >> 4, r = lane & 15;
  dst[idx] = src[(4 * ks + 2 * h + j) * C128 + nt * 16 + r];
}

// ---------------- kernel 3: k/v GEMM fused with segment stats ----------------
// one wave per 32 rows; k,v never written to HBM.
__global__ __launch_bounds__(32)
void qkv_stats_kernel(const float* __restrict__ x, const int* __restrict__ cids,
                      const float* __restrict__ WkP, const float* __restrict__ bk,
                      const float* __restrict__ WvP, const float* __restrict__ bv,
                      float* __restrict__ k_sum, float* __restrict__ v_sum,
                      unsigned* __restrict__ kmax, unsigned* __restrict__ vmax,
                      int nrows) {
  __shared__ float xs[32 * 132];
  __shared__ int   cs[32];
  const int lane = threadIdx.x;
  const int h = lane >> 4, r = lane & 15;
  const int row0 = blockIdx.x * 32;

  for (int m = 0; m < 32; ++m) {
    int row = row0 + m; if (row >= nrows) row = nrows - 1;
    *(v4f*)&xs[m * 132 + lane * 4] = *(const v4f*)(x + (size_t)row * C128 + lane * 4);
  }
  { int row = row0 + lane; if (row >= nrows) row = nrows - 1; cs[lane] = cids[row]; }
  __syncthreads();

  for (int nt = 0; nt < 8; ++nt) {
    const int col = nt * 16 + r;
    v8f ak[2], av[2];
    const float bkv = bk[col], bvv = bv[col];
    for (int mt = 0; mt < 2; ++mt)
      for (int i = 0; i < 8; ++i) { ak[mt][i] = bkv; av[mt][i] = bvv; }
    const float2* pk = (const float2*)WkP + (size_t)nt * 32 * 32;
    const float2* pv = (const float2*)WvP + (size_t)nt * 32 * 32;
    for (int ks = 0; ks < 32; ++ks) {
      float2 wk = pk[ks * 32 + lane];
      float2 wv = pv[ks * 32 + lane];
      v2f bkf = {wk.x, wk.y}, bvf = {wv.x, wv.y};
      for (int mt = 0; mt < 2; ++mt) {
        float2 a2 = *(const float2*)&xs[(mt * 16 + r) * 132 + 4 * ks + 2 * h];
        v2f af = {a2.x, a2.y};
        ak[mt] = wmma4(af, bkf, ak[mt]);
        av[mt] = wmma4(af, bvf, av[mt]);
      }
    }
    for (int mt = 0; mt < 2; ++mt)
      for (int i = 0; i < 8; ++i) {
        int m = mt * 16 + i + 8 * h;
        int row = row0 + m;
        if (row < nrows) {
          size_t off = (size_t)cs[m] * C128 + col;
          atomicAdd(&k_sum[off], ak[mt][i]);
          atomicAdd(&v_sum[off], av[mt][i]);
          atomicMax(&kmax[off], fenc(ak[mt][i]));
          atomicMax(&vmax[off], fenc(av[mt][i]));
        }
      }
  }
}

// ---------------- kernel 4: finalize centroids -------------------------------
__global__ void finalize_kernel(const float* __restrict__ k_sum,
                                const float* __restrict__ v_sum,
                                const unsigned* __restrict__ kmax,
                                const unsigned* __restrict__ vmax,
                                const unsigned* __restrict__ counts,
                                float* __restrict__ kcent,
                                float* __restrict__ vcent) {
  int idx = blockIdx.x * 256 + threadIdx.x;
  if (idx >= NBUCK * C128) return;
  int b = idx >> 7, c = idx & 127;
  unsigned cnt = counts[b];
  float cntf = (float)(cnt ? cnt : 1u);
  float km = cnt ? fdec(kmax[idx]) : 0.0f;
  float vm = cnt ? fdec(vmax[idx]) : 0.0f;
  size_t base = (size_t)b * 256 + c;
  kcent[base]       = k_sum[idx] / cntf;
  kcent[base + 128] = km;
  vcent[base]       = v_sum[idx] / cntf;
  vcent[base + 128] = vm;
}

// ---------------- kernel 5: fused q GEMM + gather + gated MLP ----------------
// one wave per 32 rows; 5 chained WMMA GEMMs with LDS ping-pong.
__global__ __launch_bounds__(32)
void fused_kernel(const float* __restrict__ x, const int* __restrict__ cids_g,
                  const float* __restrict__ WqP,  const float* __restrict__ bq,
                  const float* __restrict__ Wg1P, const float* __restrict__ bg1,
                  const float* __restrict__ Wg2P, const float* __restrict__ bg2,
                  const float* __restrict__ WvcP, const float* __restrict__ bvc,
                  const float* __restrict__ WpP,  const float* __restrict__ bp_,
                  const float* __restrict__ kcent, const float* __restrict__ vcent,
                  float* __restrict__ out, int nrows) {
  __shared__ float bufA[32 * 132];   // x -> h1 -> gate*vc
  __shared__ float bufI[32 * 260];   // interaction -> v_context
  __shared__ float bufG[32 * 132];   // gate
  __shared__ int   cs[32];
  const int lane = threadIdx.x;
  const int h = lane >> 4, r = lane & 15;
  const int row0 = blockIdx.x * 32;

  for (int m = 0; m < 32; ++m) {
    int row = row0 + m; if (row >= nrows) row = nrows - 1;
    *(v4f*)&bufA[m * 132 + lane * 4] = *(const v4f*)(x + (size_t)row * C128 + lane * 4);
  }
  { int row = row0 + lane; if (row >= nrows) row = nrows - 1; cs[lane] = cids_g[row]; }
  __syncthreads();

  // ---- stage 1: q = x@Wq + bq, interaction = concat(q,q)*k_context*scale ----
  for (int nt = 0; nt < 8; ++nt) {
    const int col = nt * 16 + r;
    v8f acc[2];
    float bb = bq[col];
    for (int mt = 0; mt < 2; ++mt) for (int i = 0; i < 8; ++i) acc[mt][i] = bb;
    const float2* wp = (const float2*)WqP + (size_t)nt * 32 * 32;
    for (int ks = 0; ks < 32; ++ks) {
      float2 w = wp[ks * 32 + lane]; v2f bf = {w.x, w.y};
      for (int mt = 0; mt < 2; ++mt) {
        float2 a2 = *(const float2*)&bufA[(mt * 16 + r) * 132 + 4 * ks + 2 * h];
        v2f af = {a2.x, a2.y};
        acc[mt] = wmma4(af, bf, acc[mt]);
      }
    }
    for (int mt = 0; mt < 2; ++mt)
      for (int i = 0; i < 8; ++i) {
        int m = mt * 16 + i + 8 * h;
        size_t cb = (size_t)cs[m] * 256 + col;
        float qv = acc[mt][i];
        bufI[m * 260 + col]       = qv * kcent[cb]       * GSCALE;
        bufI[m * 260 + 128 + col] = qv * kcent[cb + 128] * GSCALE;
      }
  }
  __syncthreads();

  // ---- stage 2: h1 = relu(inter @ Wg1 + bg1) -> bufA ----
  for (int nt = 0; nt < 8; ++nt) {
    const int col = nt * 16 + r;
    v8f acc[2];
    float bb = bg1[col];
    for (int mt = 0; mt < 2; ++mt) for (int i = 0; i < 8; ++i) acc[mt][i] = bb;
    const float2* wp = (const float2*)Wg1P + (size_t)nt * 64 * 32;
    for (int ks = 0; ks < 64; ++ks) {
      float2 w = wp[ks * 32 + lane]; v2f bf = {w.x, w.y};
      for (int mt = 0; mt < 2; ++mt) {
        float2 a2 = *(const float2*)&bufI[(mt * 16 + r) * 260 + 4 * ks + 2 * h];
        v2f af = {a2.x, a2.y};
        acc[mt] = wmma4(af, bf, acc[mt]);
      }
    }
    for (int mt = 0; mt < 2; ++mt)
      for (int i = 0; i < 8; ++i) {
        int m = mt * 16 + i + 8 * h;
        bufA[m * 132 + col] = fmaxf(acc[mt][i], 0.0f);
      }
  }
  __syncthreads();

  // ---- stage 3: gate = sigmoid(h1 @ Wg2 + bg2) -> bufG ----
  for (int nt = 0; nt < 8; ++nt) {
    const int col = nt * 16 + r;
    v8f acc[2];
    float bb = bg2[col];
    for (int mt = 0; mt < 2; ++mt) for (int i = 0; i < 8; ++i) acc[mt][i] = bb;
    const float2* wp = (const float2*)Wg2P + (size_t)nt * 32 * 32;
    for (int ks = 0; ks < 32; ++ks) {
      float2 w = wp[ks * 32 + lane]; v2f bf = {w.x, w.y};
      for (int mt = 0; mt < 2; ++mt) {
        float2 a2 = *(const float2*)&bufA[(mt * 16 + r) * 132 + 4 * ks + 2 * h];
        v2f af = {a2.x, a2.y};
        acc[mt] = wmma4(af, bf, acc[mt]);
      }
    }
    for (int mt = 0; mt < 2; ++mt)
      for (int i = 0; i < 8; ++i) {
        int m = mt * 16 + i + 8 * h;
        bufG[m * 132 + col] = 1.0f / (1.0f + expf(-acc[mt][i]));
      }
  }
  __syncthreads();

  // ---- stage 4: gather v_context into bufI (L2-hot) ----
  for (int m = 0; m < 32; ++m) {
    const float* src = vcent + (size_t)cs[m] * 256;
    *(v4f*)&bufI[m * 260 + lane * 4]       = *(const v4f*)(src + lane * 4);
    *(v4f*)&bufI[m * 260 + 128 + lane * 4] = *(const v4f*)(src + 128 + lane * 4);
  }
  __syncthreads();

  // ---- stage 5: gv = gate * (vctx @ Wvc + bvc) -> bufA ----
  for (int nt = 0; nt < 8; ++nt) {
    const int col = nt * 16 + r;
    v8f acc[2];
    float bb = bvc[col];
    for (int mt = 0; mt < 2; ++mt) for (int i = 0; i < 8; ++i) acc[mt][i] = bb;
    const float2* wp = (const float2*)WvcP + (size_t)nt * 64 * 32;
    for (int ks = 0; ks < 64; ++ks) {
      float2 w = wp[ks * 32 + lane]; v2f bf = {w.x, w.y};
      for (int mt = 0; mt < 2; ++mt) {
        float2 a2 = *(const float2*)&bufI[(mt * 16 + r) * 260 + 4 * ks + 2 * h];
        v2f af = {a2.x, a2.y};
        acc[mt] = wmma4(af, bf, acc[mt]);
      }
    }
    for (int mt = 0; mt < 2; ++mt)
      for (int i = 0; i < 8; ++i) {
        int m = mt * 16 + i + 8 * h;
        bufA[m * 132 + col] = acc[mt][i] * bufG[m * 132 + col];
      }
  }
  __syncthreads();

  // ---- stage 6: out = gv @ Wp + bp ----
  for (int nt = 0; nt < 8; ++nt) {
    const int col = nt * 16 + r;
    v8f acc[2];
    float bb = bp_[col];
    for (int mt = 0; mt < 2; ++mt) for (int i = 0; i < 8; ++i) acc[mt][i] = bb;
    const float2* wp = (const float2*)WpP + (size_t)nt * 32 * 32;
    for (int ks = 0; ks < 32; ++ks) {
      float2 w = wp[ks * 32 + lane]; v2f bf = {w.x, w.y};
      for (int mt = 0; mt < 2; ++mt) {
        float2 a2 = *(const float2*)&bufA[(mt * 16 + r) * 132 + 4 * ks + 2 * h];
        v2f af = {a2.x, a2.y};
        acc[mt] = wmma4(af, bf, acc[mt]);
      }
    }
    for (int mt = 0; mt < 2; ++mt)
      for (int i = 0; i < 8; ++i) {
        int m = mt * 16 + i + 8 * h;
        int row = row0 + m;
        if (row < nrows) out[(size_t)row * C128 + col] = acc[mt][i];
      }
  }
}

// ---------------------------------------------------------------------------
extern "C" void kernel_launch(void* const* d_in, const int* in_sizes, int n_in,
                              void* d_out, int out_size, void* d_ws, size_t ws_size,
                              hipStream_t stream) {
  const float* x   = (const float*)d_in[0];
  const int*   cid = (const int*)d_in[1];
  const float* Wq  = (const float*)d_in[3];  const float* bq  = (const float*)d_in[4];
  const float* Wk  = (const float*)d_in[5];  const float* bk  = (const float*)d_in[6];
  const float* Wv  = (const float*)d_in[7];  const float* bv  = (const float*)d_in[8];
  const float* Wg1 = (const float*)d_in[9];  const float* bg1 = (const float*)d_in[10];
  const float* Wg2 = (const float*)d_in[11]; const float* bg2 = (const float*)d_in[12];
  const float* Wvc = (const float*)d_in[13]; const float* bvc = (const float*)d_in[14];
  const float* Wp  = (const float*)d_in[15]; const float* bp  = (const float*)d_in[16];
  float* out = (float*)d_out;
  const int N = in_sizes[0] / C128;

  // workspace layout
  const size_t bc = (size_t)NBUCK * C128;
  float*    k_sum  = (float*)d_ws;
  float*    v_sum  = k_sum + bc;
  unsigned* kmax   = (unsigned*)(v_sum + bc);
  unsigned* vmax   = kmax + bc;
  unsigned* counts = vmax + bc;
  float*    kcent  = (float*)(counts + NBUCK);
  float*    vcent  = kcent + 2 * bc;
  float*    WqP    = vcent + 2 * bc;
  float*    WkP    = WqP + 128 * 128;
  float*    WvP    = WkP + 128 * 128;
  float*    Wg1P   = WvP + 128 * 128;
  float*    Wg2P   = Wg1P + 256 * 128;
  float*    WvcP   = Wg2P + 128 * 128;
  float*    WpP    = WvcP + 256 * 128;
  const size_t need = (size_t)((char*)(WpP + 128 * 128) - (char*)d_ws);
  if (ws_size < need) return;

  // 0) zero stats (sums, encoded maxes, counts) — all-zero is identity for each
  const size_t zn = 4 * bc + NBUCK;
  zero_kernel<<<(unsigned)((zn + 255) / 256), 256, 0, stream>>>((unsigned*)d_ws, zn);

  // 1) bucket counts
  counts_kernel<<<(N + 255) / 256, 256, 0, stream>>>(cid, counts, N);

  // 2) repack weights into WMMA B-fragment order
  repack_kernel<<<(128 * 128 + 255) / 256, 256, 0, stream>>>(Wq,  WqP,  128);
  repack_kernel<<<(128 * 128 + 255) / 256, 256, 0, stream>>>(Wk,  WkP,  128);
  repack_kernel<<<(128 * 128 + 255) / 256, 256, 0, stream>>>(Wv,  WvP,  128);
  repack_kernel<<<(256 * 128 + 255) / 256, 256, 0, stream>>>(Wg1, Wg1P, 256);
  repack_kernel<<<(128 * 128 + 255) / 256, 256, 0, stream>>>(Wg2, Wg2P, 128);
  repack_kernel<<<(256 * 128 + 255) / 256, 256, 0, stream>>>(Wvc, WvcP, 256);
  repack_kernel<<<(128 * 128 + 255) / 256, 256, 0, stream>>>(Wp,  WpP,  128);

  const int nb32 = (N + 31) / 32;

  // 3) k/v GEMM + segment stats (atomics into L2-resident buffers)
  qkv_stats_kernel<<<nb32, 32, 0, stream>>>(x, cid, WkP, bk, WvP, bv,
                                            k_sum, v_sum, kmax, vmax, N);

  // 4) centroids
  finalize_kernel<<<(NBUCK * C128) / 256, 256, 0, stream>>>(k_sum, v_sum, kmax,
                                                            vmax, counts, kcent,
                                                            vcent);

  // 5) fused q GEMM + gather + gated MLP
  fused_kernel<<<nb32, 32, 0, stream>>>(x, cid, WqP, bq, Wg1P, bg1, Wg2P, bg2,
                                        WvcP, bvc, WpP, bp, kcent, vcent, out, N);
}